// Forest_24962349924771
// MI455X (gfx1250) — compile-verified
//
#include <hip/hip_runtime.h>
#include <hip/hip_bf16.h>
#include <stdint.h>

// Soft decision forest: B=256, T=64, N=341 nodes (W=4, depth=4), 1024 leaves.
// Memory-bound (~156 MB total traffic -> ~6.7us floor @ 23.3 TB/s).
// Strategy: async global->LDS staging (CDNA5 ASYNCcnt path), in-place width-4
// softmax in LDS, tree prefix-product per thread, coalesced b128 stores.

#define FOREST_BATCH      256
#define FOREST_NUM_TREE   64
#define FOREST_NODES      341
#define FOREST_WIDTH      4
#define FOREST_LEAVES     1024

__device__ __forceinline__ uint32_t lds_byte_off(const void* p) {
  // generic -> LDS(as3) pointer -> 32-bit byte offset within wave's LDS alloc
  return (uint32_t)(uintptr_t)(__attribute__((address_space(3))) const void*)p;
}

__device__ __forceinline__ void async_copy_b128(uint32_t lds_off, const void* gaddr) {
  // GLOBAL_LOAD_ASYNC_TO_LDS_B128: vdst = LDS byte address, vaddr = 64-bit global
  asm volatile("global_load_async_to_lds_b128 %0, %1, off"
               :
               : "v"(lds_off), "v"((uint64_t)(uintptr_t)gaddr)
               : "memory");
}

__device__ __forceinline__ void wait_asynccnt0() {
#if __has_builtin(__builtin_amdgcn_s_wait_asynccnt)
  __builtin_amdgcn_s_wait_asynccnt(0);
#else
  asm volatile("s_wait_asynccnt 0" ::: "memory");
#endif
}

__global__ __launch_bounds__(256) void forest_kernel(const float* __restrict__ in,
                                                     float* __restrict__ out) {
  __shared__ float sm[FOREST_NODES * FOREST_WIDTH];  // 5456 bytes

  const int bt  = blockIdx.x;    // 0 .. B*T-1
  const int tid = threadIdx.x;   // 0 .. 255

  const float* src = in + (size_t)bt * (FOREST_NODES * FOREST_WIDTH);

  // ---- Stage 1: async copy 341 float4 node records into LDS ----
  {
    const uint32_t base = lds_byte_off(sm);
    async_copy_b128(base + (uint32_t)tid * 16u, src + tid * 4);
    if (tid < FOREST_NODES - 256) {  // remaining 85 records
      async_copy_b128(base + (uint32_t)(256 + tid) * 16u, src + (256 + tid) * 4);
    }
  }
  wait_asynccnt0();
  __syncthreads();

  // ---- Stage 2: in-place softmax over each width-4 node ----
  for (int n = tid; n < FOREST_NODES; n += 256) {
    float4 x = *(const float4*)&sm[n * 4];
    float m  = fmaxf(fmaxf(x.x, x.y), fmaxf(x.z, x.w));
    float e0 = expf(x.x - m);
    float e1 = expf(x.y - m);
    float e2 = expf(x.z - m);
    float e3 = expf(x.w - m);
    float r  = 1.0f / (e0 + e1 + e2 + e3);
    float4 p;
    p.x = e0 * r; p.y = e1 * r; p.z = e2 * r; p.w = e3 * r;
    *(float4*)&sm[n * 4] = p;
  }
  __syncthreads();

  // ---- Stage 3: leaf probabilities. Thread tid owns leaves [4*tid, 4*tid+4) ----
  // Leaf index = d0*256 + d1*64 + d2*16 + d3*4 + d4 (DFS lexicographic order).
  const int d0 = tid >> 6;
  const int d1 = (tid >> 4) & 3;
  const int d2 = (tid >> 2) & 3;
  const int d3 = tid & 3;
  // DFS preorder child indexing: child = i + 1 + d * s(m), s = {85, 21, 5, 1}
  const int n1 = 1 + d0 * 85;
  const int n2 = n1 + 1 + d1 * 21;
  const int n3 = n2 + 1 + d2 * 5;
  const int n4 = n3 + 1 + d3;

  const float a = sm[d0]                 // root node (node 0)
                * sm[n1 * 4 + d1]
                * sm[n2 * 4 + d2]
                * sm[n3 * 4 + d3];

  float4 p4 = *(const float4*)&sm[n4 * 4];
  float4 o;
  o.x = a * p4.x; o.y = a * p4.y; o.z = a * p4.z; o.w = a * p4.w;

  *(float4*)(out + (size_t)bt * FOREST_LEAVES + (size_t)tid * 4) = o;  // coalesced b128
}

extern "C" void kernel_launch(void* const* d_in, const int* in_sizes, int n_in,
                              void* d_out, int out_size, void* d_ws, size_t ws_size,
                              hipStream_t stream) {
  const float* in = (const float*)d_in[0];
  float* out      = (float*)d_out;
  (void)in_sizes; (void)n_in; (void)out_size; (void)d_ws; (void)ws_size;

  dim3 grid(FOREST_BATCH * FOREST_NUM_TREE);  // 16384 blocks, one per (b, t)
  dim3 block(256);                            // 8 wave32s
  hipLaunchKernelGGL(forest_kernel, grid, block, 0, stream, in, out);
}